// ResidualGCNLayer_88527865905437
// MI455X (gfx1250) — compile-verified
//
#include <hip/hip_runtime.h>
#include <stdint.h>

typedef float v2f __attribute__((ext_vector_type(2)));
typedef float v8f __attribute__((ext_vector_type(8)));

#define NN 50000
#define NE 800000
#define DD 64

// ---------------- zero init: agg (in d_out) and deg (in ws) ----------------
__global__ void gcn_zero_kernel(float* __restrict__ agg, float* __restrict__ deg,
                                int nagg, int ndeg) {
    int i = blockIdx.x * blockDim.x + threadIdx.x;
    int stride = gridDim.x * blockDim.x;
    for (int k = i; k < nagg; k += stride) agg[k] = 0.0f;
    for (int k = i; k < ndeg; k += stride) deg[k] = 0.0f;
}

// ---------------- degree: deg[col[e]] += 1 ----------------
__global__ void gcn_degree_kernel(const int* __restrict__ col, float* __restrict__ deg, int E) {
    int i = blockIdx.x * blockDim.x + threadIdx.x;
    if (i < E) atomicAdd(&deg[col[i]], 1.0f);
}

// ---------------- dinv = rsqrt(deg + 1)  (self loop folded in; always > 0) ----------------
__global__ void gcn_dinv_kernel(float* __restrict__ deg, int N) {
    int i = blockIdx.x * blockDim.x + threadIdx.x;
    if (i < N) deg[i] = rsqrtf(deg[i] + 1.0f);
}

// ---------------- h = x @ W via V_WMMA_F32_16X16X4_F32 ----------------
// One wave per 16-row strip; 4 col-tiles of 16 -> full 16x64 output per wave.
// A (16x4 f32): lane ll (0..15), half hl: VGPR0 = x[m0+ll][k0+2*hl], VGPR1 = +1
// B (4x16 f32): VGPR0 = W[k0+2*hl][n0+ll],  VGPR1 = W[k0+2*hl+1][n0+ll]
// D (16x16 f32): VGPR j = row m0 + j + 8*hl, col n0 + ll
__global__ void gcn_gemm_wmma_kernel(const float* __restrict__ x,
                                     const float* __restrict__ W,
                                     float* __restrict__ h) {
    int wave = blockIdx.x * (blockDim.x >> 5) + (threadIdx.x >> 5);
    int m0 = wave << 4;
    if (m0 >= NN) return;                 // uniform per wave: EXEC stays all-ones
    int lane = threadIdx.x & 31;
    int ll = lane & 15;
    int hl = lane >> 4;

    v8f acc0 = {}, acc1 = {}, acc2 = {}, acc3 = {};
    const float* xrow = x + (size_t)(m0 + ll) * DD;

    for (int k0 = 0; k0 < DD; k0 += 4) {
        int kb = k0 + 2 * hl;             // even -> 8B aligned float2 load
        float2 av = *(const float2*)(xrow + kb);
        v2f a; a.x = av.x; a.y = av.y;

        const float* w0 = W + (size_t)kb * DD;
        const float* w1 = w0 + DD;
        v2f b0, b1, b2, b3;
        b0.x = w0[ll];      b0.y = w1[ll];
        b1.x = w0[16 + ll]; b1.y = w1[16 + ll];
        b2.x = w0[32 + ll]; b2.y = w1[32 + ll];
        b3.x = w0[48 + ll]; b3.y = w1[48 + ll];

        acc0 = __builtin_amdgcn_wmma_f32_16x16x4_f32(false, a, false, b0, (short)0, acc0, false, false);
        acc1 = __builtin_amdgcn_wmma_f32_16x16x4_f32(false, a, false, b1, (short)0, acc1, false, false);
        acc2 = __builtin_amdgcn_wmma_f32_16x16x4_f32(false, a, false, b2, (short)0, acc2, false, false);
        acc3 = __builtin_amdgcn_wmma_f32_16x16x4_f32(false, a, false, b3, (short)0, acc3, false, false);
    }

#pragma unroll
    for (int j = 0; j < 8; ++j) {
        int row = m0 + j + 8 * hl;
        float* hr = h + (size_t)row * DD + ll;
        hr[0]  = acc0[j];
        hr[16] = acc1[j];
        hr[32] = acc2[j];
        hr[48] = acc3[j];
    }
}

// ---------------- edge scatter: agg[col] += dinv[row]*dinv[col] * h[row] ----------------
// One wave per edge, float2 per lane (32 lanes x 2 = 64 channels).
__global__ void gcn_scatter_kernel(const int* __restrict__ row,
                                   const int* __restrict__ col,
                                   const float* __restrict__ dinv,
                                   const float* __restrict__ h,
                                   float* __restrict__ agg, int E) {
    int gid = blockIdx.x * blockDim.x + threadIdx.x;
    int e = gid >> 5;
    if (e >= E) return;
    int lane = gid & 31;
    int r = row[e];
    int c = col[e];
    float w = dinv[r] * dinv[c];
    float2 v = *(const float2*)(h + (size_t)r * DD + lane * 2);
    float* a = agg + (size_t)c * DD + lane * 2;
    atomicAdd(a,     w * v.x);
    atomicAdd(a + 1, w * v.y);
}

// ---------------- epilogue: out = relu(agg + dinv^2*h + b) + x ----------------
__global__ void gcn_final_kernel(const float* __restrict__ x,
                                 const float* __restrict__ h,
                                 const float* __restrict__ dinv,
                                 const float* __restrict__ b,
                                 float* __restrict__ out, int total) {
    int i = blockIdx.x * blockDim.x + threadIdx.x;
    if (i >= total) return;
    int node = i >> 6;
    int d = i & 63;
    float di = dinv[node];
    float v = out[i] + di * di * h[i] + b[d];
    out[i] = fmaxf(v, 0.0f) + x[i];
}

extern "C" void kernel_launch(void* const* d_in, const int* in_sizes, int n_in,
                              void* d_out, int out_size, void* d_ws, size_t ws_size,
                              hipStream_t stream) {
    const float* x    = (const float*)d_in[0];
    const int*   ei   = (const int*)d_in[1];   // [2, NE]: row = ei[0..E), col = ei[E..2E)
    const float* W    = (const float*)d_in[2];
    const float* bias = (const float*)d_in[3];

    float* out  = (float*)d_out;               // doubles as agg accumulator
    float* dinv = (float*)d_ws;                // NN floats
    float* h    = dinv + NN;                   // NN*DD floats

    const int* row = ei;
    const int* col = ei + NE;

    // 1) zero agg + deg (deterministic each call)
    gcn_zero_kernel<<<2048, 256, 0, stream>>>(out, dinv, NN * DD, NN);

    // 2) degree accumulation
    gcn_degree_kernel<<<(NE + 255) / 256, 256, 0, stream>>>(col, dinv, NE);

    // 3) dinv = rsqrt(deg + 1)
    gcn_dinv_kernel<<<(NN + 255) / 256, 256, 0, stream>>>(dinv, NN);

    // 4) h = x @ W  (WMMA): 3125 row tiles, 8 waves/block
    int ntiles = NN / 16;                      // 3125, exact
    gcn_gemm_wmma_kernel<<<(ntiles + 7) / 8, 256, 0, stream>>>(x, W, h);

    // 5) edge scatter (1 wave per edge)
    gcn_scatter_kernel<<<(NE * 32 + 255) / 256, 256, 0, stream>>>(row, col, dinv, h, out, NE);

    // 6) epilogue
    gcn_final_kernel<<<(NN * DD + 255) / 256, 256, 0, stream>>>(x, h, dinv, bias, out, NN * DD);
}